// MPNEncoder_21844203667544
// MI455X (gfx1250) — compile-verified
//
#include <hip/hip_runtime.h>

// chemprop D-MPNN encoder for MI455X (gfx1250), fp32 end-to-end using
// V_WMMA_F32_16X16X4_F32. Memory-bound workload -> fuse gathers/epilogues,
// keep full fp32 fidelity. Weights with odd K are pre-padded into workspace
// so every WMMA inner loop is branch-free (aligned b64 fragment loads only).

#define N_ATOMS_C   100000
#define N_BONDS_C   200000
#define MAX_NB_C    6
#define ATOM_FDIM_C 133
#define BOND_FDIM_C 147
#define HIDDEN_C    256
#define DEPTH_C     5

#define KI_PAD 148   // 147 -> 148
#define KO_PAD 392   // 389 -> 392
#define KO_RAW (ATOM_FDIM_C + HIDDEN_C)   // 389

typedef __attribute__((ext_vector_type(2))) float v2f;
typedef __attribute__((ext_vector_type(8))) float v8f;

// ---------------------------------------------------------------------------
// Kernel 0: zero-pad W_i and W_o along K into workspace (run once per launch).
// ---------------------------------------------------------------------------
__global__ __launch_bounds__(256) void pad_weights_kernel(
    const float* __restrict__ W_i, const float* __restrict__ W_o,
    float* __restrict__ Wi_pad, float* __restrict__ Wo_pad)
{
    const int gid = blockIdx.x * 256 + threadIdx.x;
    const int NI = HIDDEN_C * KI_PAD;                   // 37888
    const int NO = HIDDEN_C * KO_PAD;                   // 100352
    if (gid < NI) {
        const int n = gid / KI_PAD, k = gid - n * KI_PAD;
        Wi_pad[gid] = (k < BOND_FDIM_C) ? W_i[(size_t)n * BOND_FDIM_C + k] : 0.f;
    } else if (gid < NI + NO) {
        const int g2 = gid - NI;
        const int n = g2 / KO_PAD, k = g2 - n * KO_PAD;
        Wo_pad[g2] = (k < KO_RAW) ? W_o[(size_t)n * KO_RAW + k] : 0.f;
    }
}

// ---------------------------------------------------------------------------
// Shared wave-level WMMA micro-kernel: 16x128 output per wave, K-step = 4.
// A fragments come from LDS (padded, 8B aligned); B fragments from a padded
// global weight matrix (row stride = kpad, even). Branch-free inner loop:
// 8 b64 loads batched, then 8 WMMAs.
// ---------------------------------------------------------------------------
__device__ __forceinline__ void wave_gemm(
    const float* __restrict__ Asmem, int astride, int arow, int half,
    const float* __restrict__ Wpad, int kpad, int colbase, int l16,
    v8f acc[8])
{
    for (int kb = 0; kb < kpad; kb += 4) {
        const int kk = kb + 2 * half;                   // this lane's K pair
        const v2f a = *reinterpret_cast<const v2f*>(Asmem + (size_t)arow * astride + kk);
        v2f bf[8];
        #pragma unroll
        for (int t = 0; t < 8; ++t) {
            const int n = colbase + t * 16 + l16;
            bf[t] = *reinterpret_cast<const v2f*>(Wpad + (size_t)n * kpad + kk);
        }
        #pragma unroll
        for (int t = 0; t < 8; ++t) {
            acc[t] = __builtin_amdgcn_wmma_f32_16x16x4_f32(
                false, a, false, bf[t], (short)0, acc[t], false, false);
        }
    }
}

// ---------------------------------------------------------------------------
// Kernel 1: inp = f_bonds @ W_i^T ; msg = relu(inp)
// 256 threads (8 waves). Block tile: 64 bonds x 256 hidden. K padded to 148.
// ---------------------------------------------------------------------------
__global__ __launch_bounds__(256) void wi_gemm_kernel(
    const float* __restrict__ f_bonds, const float* __restrict__ Wi_pad,
    float* __restrict__ inp, float* __restrict__ msg)
{
    __shared__ float As[64][KI_PAD];                    // stride even -> aligned v2f
    const int rb  = blockIdx.x * 64;
    const int tid = threadIdx.x;
    for (int idx = tid; idx < 64 * KI_PAD; idx += 256) {
        const int r = idx / KI_PAD, k = idx - r * KI_PAD;
        As[r][k] = (k < BOND_FDIM_C) ? f_bonds[(size_t)(rb + r) * BOND_FDIM_C + k] : 0.f;
    }
    __syncthreads();

    const int lane = tid & 31, wid = tid >> 5;
    const int mrow    = (wid & 3) * 16;
    const int colbase = (wid >> 2) * 128;
    const int l16 = lane & 15, half = lane >> 4;

    const v8f vzero = {};
    v8f acc[8];
    #pragma unroll
    for (int t = 0; t < 8; ++t) acc[t] = vzero;

    wave_gemm(&As[0][0], KI_PAD, mrow + l16, half, Wi_pad, KI_PAD, colbase, l16, acc);

    #pragma unroll
    for (int t = 0; t < 8; ++t) {
        const int g = colbase + t * 16 + l16;
        #pragma unroll
        for (int v = 0; v < 8; ++v) {
            const int bm = rb + mrow + v + 8 * half;    // C layout: lanes>=16 hold M=v+8
            const size_t o = (size_t)bm * HIDDEN_C + g;
            const float val = acc[t][v];
            inp[o] = val;
            msg[o] = val > 0.f ? val : 0.f;
        }
    }
}

// ---------------------------------------------------------------------------
// Kernel 2: a_message[a] = sum_j message[a2b[a][j]]   (memory-bound gather)
// ---------------------------------------------------------------------------
__global__ __launch_bounds__(256) void gather_kernel(
    const float* __restrict__ msg, const int* __restrict__ a2b,
    float* __restrict__ amsg)
{
    const int gid = blockIdx.x * 256 + threadIdx.x;     // N_ATOMS * (HIDDEN/4) threads
    const int a   = gid >> 6;
    const int c4  = (gid & 63) << 2;
    float4 s = make_float4(0.f, 0.f, 0.f, 0.f);
    #pragma unroll
    for (int j = 0; j < MAX_NB_C; ++j) {
        const int bi = a2b[a * MAX_NB_C + j];
        const float4 m = *reinterpret_cast<const float4*>(msg + (size_t)bi * HIDDEN_C + c4);
        s.x += m.x; s.y += m.y; s.z += m.z; s.w += m.w;
    }
    *reinterpret_cast<float4*>(amsg + (size_t)a * HIDDEN_C + c4) = s;
}

// ---------------------------------------------------------------------------
// Kernel 3: msg_out = relu(inp + (amsg[b2a] - msg_in[b2revb]) @ W_h^T)
// Gather of m fused into the A-tile LDS staging (float4). K=256.
// ---------------------------------------------------------------------------
__global__ __launch_bounds__(256) void wh_gemm_kernel(
    const float* __restrict__ inp, const float* __restrict__ msg_in,
    const float* __restrict__ amsg, const int* __restrict__ b2a,
    const int* __restrict__ b2revb, const float* __restrict__ W_h,
    float* __restrict__ msg_out)
{
    __shared__ float Ms[64][HIDDEN_C];                  // 64 KB
    __shared__ int   sia[64], sir[64];
    const int rb  = blockIdx.x * 64;
    const int tid = threadIdx.x;
    if (tid < 64)       sia[tid] = b2a[rb + tid];
    else if (tid < 128) sir[tid - 64] = b2revb[rb + tid - 64];
    __syncthreads();
    // stage m = amsg[b2a] - msg_in[b2revb], float4-vectorized
    for (int idx = tid; idx < 64 * (HIDDEN_C / 4); idx += 256) {
        const int r  = idx >> 6;
        const int c4 = (idx & 63) << 2;
        const float4 ga = *reinterpret_cast<const float4*>(amsg   + (size_t)sia[r] * HIDDEN_C + c4);
        const float4 gr = *reinterpret_cast<const float4*>(msg_in + (size_t)sir[r] * HIDDEN_C + c4);
        float4 m;
        m.x = ga.x - gr.x; m.y = ga.y - gr.y; m.z = ga.z - gr.z; m.w = ga.w - gr.w;
        *reinterpret_cast<float4*>(&Ms[r][c4]) = m;
    }
    __syncthreads();

    const int lane = tid & 31, wid = tid >> 5;
    const int mrow    = (wid & 3) * 16;
    const int colbase = (wid >> 2) * 128;
    const int l16 = lane & 15, half = lane >> 4;

    const v8f vzero = {};
    v8f acc[8];
    #pragma unroll
    for (int t = 0; t < 8; ++t) acc[t] = vzero;

    wave_gemm(&Ms[0][0], HIDDEN_C, mrow + l16, half, W_h, HIDDEN_C, colbase, l16, acc);

    #pragma unroll
    for (int t = 0; t < 8; ++t) {
        const int g = colbase + t * 16 + l16;
        #pragma unroll
        for (int v = 0; v < 8; ++v) {
            const int bm = rb + mrow + v + 8 * half;
            const size_t o = (size_t)bm * HIDDEN_C + g;
            const float val = inp[o] + acc[t][v];
            msg_out[o] = val > 0.f ? val : 0.f;
        }
    }
}

// ---------------------------------------------------------------------------
// Kernel 4: out = relu(concat(f_atoms, amsg) @ W_o^T + b_o)
// K = 133 + 256 = 389, pad -> 392. Concat fused into LDS staging.
// ---------------------------------------------------------------------------
__global__ __launch_bounds__(256) void wo_gemm_kernel(
    const float* __restrict__ f_atoms, const float* __restrict__ amsg,
    const float* __restrict__ Wo_pad, const float* __restrict__ b_o,
    float* __restrict__ out)
{
    __shared__ float As[64][KO_PAD];                    // ~100 KB
    const int rb  = blockIdx.x * 64;
    const int tid = threadIdx.x;
    for (int idx = tid; idx < 64 * KO_PAD; idx += 256) {
        const int r = idx / KO_PAD, k = idx - r * KO_PAD;
        const int a = rb + r;
        float v = 0.f;
        if (a < N_ATOMS_C) {
            if (k < ATOM_FDIM_C)  v = f_atoms[(size_t)a * ATOM_FDIM_C + k];
            else if (k < KO_RAW)  v = amsg[(size_t)a * HIDDEN_C + (k - ATOM_FDIM_C)];
        }
        As[r][k] = v;
    }
    __syncthreads();

    const int lane = tid & 31, wid = tid >> 5;
    const int mrow    = (wid & 3) * 16;
    const int colbase = (wid >> 2) * 128;
    const int l16 = lane & 15, half = lane >> 4;

    const v8f vzero = {};
    v8f acc[8];
    #pragma unroll
    for (int t = 0; t < 8; ++t) acc[t] = vzero;

    wave_gemm(&As[0][0], KO_PAD, mrow + l16, half, Wo_pad, KO_PAD, colbase, l16, acc);

    #pragma unroll
    for (int t = 0; t < 8; ++t) {
        const int g = colbase + t * 16 + l16;
        const float bias = b_o[g];
        #pragma unroll
        for (int v = 0; v < 8; ++v) {
            const int am = rb + mrow + v + 8 * half;
            if (am < N_ATOMS_C) {
                const float val = acc[t][v] + bias;
                out[(size_t)am * HIDDEN_C + g] = val > 0.f ? val : 0.f;
            }
        }
    }
}

// ---------------------------------------------------------------------------
extern "C" void kernel_launch(void* const* d_in, const int* in_sizes, int n_in,
                              void* d_out, int out_size, void* d_ws, size_t ws_size,
                              hipStream_t stream)
{
    const float* f_atoms = (const float*)d_in[0];
    const float* f_bonds = (const float*)d_in[1];
    const int*   a2b     = (const int*)d_in[2];
    const int*   b2a     = (const int*)d_in[3];
    const int*   b2revb  = (const int*)d_in[4];
    const float* W_i     = (const float*)d_in[5];
    const float* W_h     = (const float*)d_in[6];
    const float* W_o     = (const float*)d_in[7];
    const float* b_o     = (const float*)d_in[8];
    float* out = (float*)d_out;

    const size_t SZ = (size_t)N_BONDS_C * HIDDEN_C;     // 51.2M floats
    const size_t AZ = (size_t)N_ATOMS_C * HIDDEN_C;     // 25.6M floats
    float* ws     = (float*)d_ws;
    float* inp    = ws;                                 // [n_bonds, H] pre-activation
    float* msgA   = ws + SZ;                            // message double-buffer
    float* msgB   = ws + 2 * SZ;
    float* amsg   = ws + 3 * SZ;                        // [n_atoms, H]
    float* Wi_pad = ws + 3 * SZ + AZ;                   // [256, 148]
    float* Wo_pad = Wi_pad + (size_t)HIDDEN_C * KI_PAD; // [256, 392]

    const dim3 blk(256);
    const int pad_blocks       = (HIDDEN_C * (KI_PAD + KO_PAD) + 255) / 256;
    const int gemm_bond_blocks = N_BONDS_C / 64;                     // 3125
    const int gather_blocks    = (N_ATOMS_C * (HIDDEN_C / 4)) / 256; // 25000
    const int gemm_atom_blocks = (N_ATOMS_C + 63) / 64;              // 1563

    pad_weights_kernel<<<pad_blocks, blk, 0, stream>>>(W_i, W_o, Wi_pad, Wo_pad);
    wi_gemm_kernel<<<gemm_bond_blocks, blk, 0, stream>>>(f_bonds, Wi_pad, inp, msgA);
    for (int d = 0; d < DEPTH_C - 1; ++d) {
        gather_kernel<<<gather_blocks, blk, 0, stream>>>(msgA, a2b, amsg);
        wh_gemm_kernel<<<gemm_bond_blocks, blk, 0, stream>>>(
            inp, msgA, amsg, b2a, b2revb, W_h, msgB);
        float* t = msgA; msgA = msgB; msgB = t;
    }
    gather_kernel<<<gather_blocks, blk, 0, stream>>>(msgA, a2b, amsg);
    wo_gemm_kernel<<<gemm_atom_blocks, blk, 0, stream>>>(f_atoms, amsg, W_o 
        == nullptr ? Wo_pad : Wo_pad, b_o, out);
}